// TDVPFlexRNNCell_84937273245775
// MI455X (gfx1250) — compile-verified
//
#include <hip/hip_runtime.h>

// ---------------- static problem configuration (mirrors reference) ---------
#define PDIM      2
#define DDIM      10          // SDE state dim
#define NUM_TRAJ  128
#define BATCH     64
#define NTRAJ_TOT (NUM_TRAJ * BATCH)   // 8192
#define NSTEPS    255
#define DT        0.00390625f          // 2^-8

typedef float v2f __attribute__((ext_vector_type(2)));
typedef float v8f __attribute__((ext_vector_type(8)));

// swap lane halves (lane L <-> lane L^16) within a wave32.
// ds_swizzle group-of-32 encoding: offset[14:10]=xor, [9:5]=or, [4:0]=and
//   SWAPX16: xor=0x10, or=0x00, and=0x1f  ->  0x401F
__device__ __forceinline__ float swap16(float x) {
    return __int_as_float(__builtin_amdgcn_ds_swizzle(__float_as_int(x), 0x401F));
}

// ---------------------------------------------------------------------------
// Kernel 1: Euler-Maruyama scan. One wave32 integrates 16 trajectories.
// State kept in WMMA C/D layout: S[v], lanes 0-15 -> (comp v,  traj=lane),
//                                       lanes16-31 -> (comp v+8, traj=lane-16)
// Weight matrices kept in WMMA A-operand layout (M=out comp, K=in comp);
// x^T rebuilt into B-operand layout each step with 4 packed half-swaps.
// ---------------------------------------------------------------------------
__global__ __launch_bounds__(256) void sde_scan_kernel(
    const float* __restrict__ inputs,   // [64]
    const float* __restrict__ rho0,     // [64,2,2]
    const float* __restrict__ theta,    // [4]
    const float* __restrict__ A,        // [10,10] row-major
    const float* __restrict__ B1,       // [10,10]
    const float* __restrict__ B2,       // [10,10]
    const float* __restrict__ dW,       // [8192,255,2]
    float*       __restrict__ xT)       // [8192,4] comps 0..3 at final time
{
    const int tid  = blockIdx.x * blockDim.x + threadIdx.x;
    const int wave = tid >> 5;
    const int lane = threadIdx.x & 31;
    const int hi   = lane >> 4;          // lane half
    const int t16  = lane & 15;          // trajectory slot within wave
    const int traj = wave * 16 + t16;
    const int b    = traj & (BATCH - 1); // sample index (tile pattern n=c*64+b)

    // ---- load weights into A-operand layout: lane(M=t16), K = 4j + 2*hi + r
    v2f wA[3], wB1[3], wB2[3];
    #pragma unroll
    for (int j = 0; j < 3; ++j) {
        #pragma unroll
        for (int r = 0; r < 2; ++r) {
            const int k  = 4 * j + 2 * hi + r;
            const bool ok = (t16 < DDIM) && (k < DDIM);
            const int idx = t16 * DDIM + k;
            wA [j][r] = ok ? A [idx] : 0.0f;
            wB1[j][r] = ok ? B1[idx] : 0.0f;
            wB2[j][r] = ok ? B2[idx] : 0.0f;
        }
    }

    // ---- per-lane loop-invariant scalars
    const float th1 = theta[1], th2 = theta[2], th3 = theta[3];
    const float c1  = 1.0f + DT * th1;               // (1 + p1*dt)
    const float cA  = DT * (inputs[b] + 1e-8f);      // p0*dt  (p0 per sample)

    // ---- initial state (rho00, Re rho01, 0, rho11, zeros...)
    float S[8];
    #pragma unroll
    for (int i = 0; i < 8; ++i) S[i] = 0.0f;
    if (hi == 0) {
        S[0] = rho0[b * 4 + 0];
        S[1] = rho0[b * 4 + 1];
        S[3] = rho0[b * 4 + 3];
    }

    const float* dwp = dW + (size_t)traj * (NSTEPS * 2);

    for (int step = 0; step < NSTEPS; ++step) {
        const float2 dw = *(const float2*)(dwp + step * 2);
        const int pf = (step + 1 < NSTEPS) ? (step + 1) : step;   // branchless clamp
        __builtin_prefetch(dwp + pf * 2, 0, 0);                   // global_prefetch_b8
        const float cb1 = th2 * dw.x;   // p2 * dW0
        const float cb2 = th3 * dw.y;   // p3 * dW1

        // ---- build B-operand (x^T) chunks: chunk j, K = 4j + 2*hi + r
        // Packed swaps: one cross-half shuffle serves a hi-consumer and a
        // lo-consumer at once:  r1 = swap(hi ? S0 : S2)
        //   lo lanes: r1 = S0[partner] = comp 8 of own traj   (feeds bb2 lo)
        //   hi lanes: r1 = S2[partner] = comp 2 of own traj   (feeds bb0 hi)
        const float r1 = swap16(hi ? S[0] : S[2]);
        const float r2 = swap16(hi ? S[1] : S[3]);
        const float r3 = swap16(S[6]);   // comp 6 -> hi lanes of bb1[0]
        const float r4 = swap16(S[7]);   // comp 7 -> hi lanes of bb1[1]
        v2f bb0, bb1, bb2;
        bb0[0] = hi ? r1   : S[0];   // comps 0 / 2
        bb0[1] = hi ? r2   : S[1];   // comps 1 / 3
        bb1[0] = hi ? r3   : S[4];   // comps 4 / 6
        bb1[1] = hi ? r4   : S[5];   // comps 5 / 7
        bb2[0] = hi ? 0.0f : r1;     // comps 8 / 10(pad)
        bb2[1] = hi ? 0.0f : r2;     // comps 9 / 11(pad)

        // ---- three mat-vecs via f32 WMMA (K accumulated over 3 chunks)
        const v8f z = {};
        v8f yA = __builtin_amdgcn_wmma_f32_16x16x4_f32(false, wA[0], false, bb0, (short)0, z,  false, false);
        yA     = __builtin_amdgcn_wmma_f32_16x16x4_f32(false, wA[1], false, bb1, (short)0, yA, false, false);
        yA     = __builtin_amdgcn_wmma_f32_16x16x4_f32(false, wA[2], false, bb2, (short)0, yA, false, false);

        v8f y1 = __builtin_amdgcn_wmma_f32_16x16x4_f32(false, wB1[0], false, bb0, (short)0, z,  false, false);
        y1     = __builtin_amdgcn_wmma_f32_16x16x4_f32(false, wB1[1], false, bb1, (short)0, y1, false, false);
        y1     = __builtin_amdgcn_wmma_f32_16x16x4_f32(false, wB1[2], false, bb2, (short)0, y1, false, false);

        v8f y2 = __builtin_amdgcn_wmma_f32_16x16x4_f32(false, wB2[0], false, bb0, (short)0, z,  false, false);
        y2     = __builtin_amdgcn_wmma_f32_16x16x4_f32(false, wB2[1], false, bb1, (short)0, y2, false, false);
        y2     = __builtin_amdgcn_wmma_f32_16x16x4_f32(false, wB2[2], false, bb2, (short)0, y2, false, false);

        // ---- elementwise SDE update:
        //      x = x*(1+p1*dt) + yA*(p0*dt) + y1*(p2*dW0) + y2*(p3*dW1)
        #pragma unroll
        for (int i = 0; i < 8; ++i) {
            float s = S[i] * c1;
            s = fmaf(yA[i], cA,  s);
            s = fmaf(y1[i], cb1, s);
            s = fmaf(y2[i], cb2, s);
            S[i] = s;
        }
    }

    // ---- emit comps 0..3 (rho00, Re01, Im01, rho11) for this trajectory
    if (hi == 0) {
        float4 o = make_float4(S[0], S[1], S[2], S[3]);
        *(float4*)(xT + (size_t)traj * 4) = o;
    }
}

// ---------------------------------------------------------------------------
// Kernel 2: deterministic trajectory mean + trace norm + closed-form 2x2 eigh
// projection + outputs. One thread per sample.
// ---------------------------------------------------------------------------
__global__ __launch_bounds__(64) void finalize_kernel(
    const float* __restrict__ xT,      // [8192,4]
    const float* __restrict__ inputs,  // [64]
    float*       __restrict__ out)     // [64*3] probs||input, then [64*8] rho_proj (re,im)
{
    const int b = threadIdx.x;
    if (b >= BATCH) return;

    float m0 = 0.f, m1 = 0.f, m2 = 0.f, m3 = 0.f;
    for (int c = 0; c < NUM_TRAJ; ++c) {            // fixed order -> deterministic
        const float4 v = *(const float4*)(xT + ((size_t)(c * BATCH + b)) * 4);
        m0 += v.x; m1 += v.y; m2 += v.z; m3 += v.w;
    }
    const float inv = 1.0f / (float)NUM_TRAJ;
    m0 *= inv; m1 *= inv; m2 *= inv; m3 *= inv;

    // trace-normalized Hermitian H = [[a, br+i*bi],[br-i*bi, d]]
    const float tr = m0 + m3;
    const float a  = m0 / tr, d = m3 / tr;
    const float br = m1 / tr, bi = m2 / tr;

    // closed-form eigh of 2x2 Hermitian; eigenvalues ascending (l1 <= l2)
    const float mean = 0.5f * (a + d);
    const float disc = sqrtf(0.25f * (a - d) * (a - d) + br * br + bi * bi);
    const float l1 = mean - disc, l2 = mean + disc;
    float w1 = fmaxf(l1, 0.0f), w2 = fmaxf(l2, 0.0f);
    const float ws = w1 + w2;
    w1 /= ws; w2 /= ws;
    // rho_proj = w1*P1 + w2*P2 = (w1 - cc*l1)*I + cc*H  with P2=(H-l1*I)/(l2-l1)
    const float cc    = (w2 - w1) / (2.0f * fmaxf(disc, 1e-30f));
    const float alpha = w1 - cc * l1;
    const float r00   = alpha + cc * a;
    const float r11   = alpha + cc * d;
    const float o01r  = cc * br;
    const float o01i  = cc * bi;

    // output 0: probs (real diag) || input
    out[b * 3 + 0] = r00;
    out[b * 3 + 1] = r11;
    out[b * 3 + 2] = inputs[b];
    // output 1: rho_proj as complex64 (re,im interleaved), row-major 2x2
    float* rp = out + BATCH * 3 + b * 8;
    rp[0] = r00;  rp[1] = 0.0f;
    rp[2] = o01r; rp[3] = o01i;
    rp[4] = o01r; rp[5] = -o01i;
    rp[6] = r11;  rp[7] = 0.0f;
}

// ---------------------------------------------------------------------------
extern "C" void kernel_launch(void* const* d_in, const int* in_sizes, int n_in,
                              void* d_out, int out_size, void* d_ws, size_t ws_size,
                              hipStream_t stream) {
    const float* inputs = (const float*)d_in[0];  // [64,1]
    const float* rho0   = (const float*)d_in[1];  // [64,2,2]
    const float* theta  = (const float*)d_in[2];  // [4]
    const float* A      = (const float*)d_in[3];  // [10,10]
    const float* B1     = (const float*)d_in[4];  // [10,10]
    const float* B2     = (const float*)d_in[5];  // [10,10]
    const float* dW     = (const float*)d_in[6];  // [8192,255,2]
    float* out = (float*)d_out;
    float* xT  = (float*)d_ws;                    // 8192*4 floats = 128 KB scratch

    // 8192 trajectories / 16 per wave = 512 waves; 8 waves per 256-thread block
    sde_scan_kernel<<<64, 256, 0, stream>>>(inputs, rho0, theta, A, B1, B2, dW, xT);
    finalize_kernel<<<1, 64, 0, stream>>>(xT, inputs, out);
}